// MLGraphConstruction_48163763257589
// MI455X (gfx1250) — compile-verified
//
#include <hip/hip_runtime.h>
#include <hip/hip_bf16.h>
#include <math.h>

#define NN   16384
#define KSEL 64
#define NK   (NN * KSEL)                 // 1048576 edges

typedef float v2f __attribute__((ext_vector_type(2)));
typedef float v8f __attribute__((ext_vector_type(8)));

// output layout (floats), concatenated in reference return order
#define OFF_XCAT  ((size_t)0)
#define OFF_EI    ((size_t)NN * 22)                 // 360448
#define OFF_Y     (OFF_EI + (size_t)2 * NK)         // 2457600
#define OFF_EF    (OFF_Y + (size_t)NK)              // 3506176
#define OFF_MASK  (OFF_EF + (size_t)NK * 44)        // 49643520

static __device__ __forceinline__ v8f wmma_f32(v2f a, v2f b, v8f c) {
    return __builtin_amdgcn_wmma_f32_16x16x4_f32(false, a, false, b, (short)0, c,
                                                 false, false);
}

// ---------------------------------------------------------------------------
// Kernel A: MLP embed + L2 normalize. Writes H [N,8] (ws) and x_cat [N,22] (out)
// ---------------------------------------------------------------------------
__global__ __launch_bounds__(256) void embed_kernel(
    const float* __restrict__ x, const float* __restrict__ W1,
    const float* __restrict__ b1, const float* __restrict__ W2,
    const float* __restrict__ b2, float* __restrict__ H,
    float* __restrict__ out)
{
    int i = blockIdx.x * 256 + threadIdx.x;
    if (i >= NN) return;

    float xi[14];
#pragma unroll
    for (int k = 0; k < 14; ++k) xi[k] = x[(size_t)i * 14 + k];

    float acc[8];
#pragma unroll
    for (int c = 0; c < 8; ++c) acc[c] = b2[c];

    for (int j = 0; j < 128; ++j) {
        float a = b1[j];
#pragma unroll
        for (int k = 0; k < 14; ++k) a = fmaf(xi[k], W1[k * 128 + j], a);
        a = fmaxf(a, 0.0f);
#pragma unroll
        for (int c = 0; c < 8; ++c) acc[c] = fmaf(a, W2[j * 8 + c], acc[c]);
    }

    float n2 = 0.0f;
#pragma unroll
    for (int c = 0; c < 8; ++c) n2 = fmaf(acc[c], acc[c], n2);
    float inv = 1.0f / (sqrtf(n2) + 1e-12f);

    float* xc = out + OFF_XCAT + (size_t)i * 22;
#pragma unroll
    for (int c = 0; c < 8; ++c) {
        float h = acc[c] * inv;
        H[(size_t)i * 8 + c] = h;
        xc[c] = h;
    }
#pragma unroll
    for (int k = 0; k < 14; ++k) xc[8 + k] = xi[k];
}

// ---------------------------------------------------------------------------
// Kernel B: WMMA-fp32 dot tiles + per-row radix select of top-64 dots.
// One workgroup = 16 query rows (one WMMA M-tile), 8 waves split 1024 column
// tiles. 3 passes over the (recomputed) tile stream:
//   pass 0: 256-bin histogram on key[31:24]   (key = descending-order uint of dot)
//   pass 1: 256-bin histogram on key[23:16] inside the coarse threshold bin
//   pass 2: collect definite winners + boundary ties
// Then per-row sort descending (matches top_k's sorted output order).
// ---------------------------------------------------------------------------
__global__ __launch_bounds__(256) void knn_kernel(
    const float* __restrict__ H, int* __restrict__ nbr, float* __restrict__ dotv)
{
    __shared__ unsigned int hist[16][256];
    __shared__ unsigned int cb[16], abv0[16], th16[16], cntA[16], cntB[16];
    __shared__ int   outIdx[16][KSEL];
    __shared__ float outVal[16][KSEL];
    __shared__ int   bIdx[16][KSEL];
    __shared__ float bVal[16][KSEL];

    const int tid  = threadIdx.x;
    const int lane = tid & 31;
    const int wave = tid >> 5;
    const int i0   = blockIdx.x * 16;
    const int mrow = lane & 15;
    const int kb   = (lane < 16) ? 0 : 2;   // lanes 0-15: K=0,1; lanes 16-31: K=2,3
    const int mAdd = (lane < 16) ? 0 : 8;   // C/D layout: rows r / r+8

    for (int t = tid; t < 16 * 256; t += 256) ((unsigned int*)hist)[t] = 0u;
    if (tid < 16) { cntA[tid] = 0u; cntB[tid] = 0u; }
    __syncthreads();

    // A tile: 16 query rows, K-steps {0..3} and {4..7}
    v2f a0 = *(const v2f*)(H + (size_t)(i0 + mrow) * 8 + kb);
    v2f a1 = *(const v2f*)(H + (size_t)(i0 + mrow) * 8 + 4 + kb);

    for (int pass = 0; pass < 3; ++pass) {
        for (int t = wave; t < NN / 16; t += 8) {
            const int j0 = t * 16;
            const int n  = lane & 15;
            if (t + 8 < NN / 16)
                __builtin_prefetch(H + (size_t)((t + 8) * 16 + n) * 8 + kb, 0, 3);
            v2f b0 = *(const v2f*)(H + (size_t)(j0 + n) * 8 + kb);
            v2f b1 = *(const v2f*)(H + (size_t)(j0 + n) * 8 + 4 + kb);
            v8f c = {};
            c = wmma_f32(a0, b0, c);   // K = 0..3
            c = wmma_f32(a1, b1, c);   // K = 4..7
#pragma unroll
            for (int r = 0; r < 8; ++r) {
                const float d = c[r];
                const int M = r + mAdd;
                const int j = j0 + n;
                if (j == i0 + M) continue;         // exclude self-loop
                const unsigned u  = __float_as_uint(d);
                const unsigned ou = (u & 0x80000000u) ? ~u : (u | 0x80000000u);
                const unsigned key = ~ou;          // smaller key == larger dot
                if (pass == 0) {
                    atomicAdd(&hist[M][key >> 24], 1u);
                } else if (pass == 1) {
                    if ((key >> 24) == cb[M])
                        atomicAdd(&hist[M][(key >> 16) & 255u], 1u);
                } else {
                    const unsigned k16 = key >> 16;
                    const unsigned th  = th16[M];
                    if (k16 < th) {
                        unsigned p = atomicAdd(&cntA[M], 1u);   // p < 64 guaranteed
                        outIdx[M][p] = j; outVal[M][p] = d;
                    } else if (k16 == th) {
                        unsigned p = atomicAdd(&cntB[M], 1u);
                        if (p < KSEL) { bIdx[M][p] = j; bVal[M][p] = d; }
                    }
                }
            }
        }
        __syncthreads();
        if (pass == 0) {
            if (tid < 16) {
                unsigned c = 0, b = 0;
                for (; b < 256; ++b) {
                    unsigned nc = c + hist[tid][b];
                    if (nc >= (unsigned)KSEL) break;
                    c = nc;
                }
                cb[tid] = b; abv0[tid] = c;
            }
            __syncthreads();
            for (int t = tid; t < 16 * 256; t += 256) ((unsigned int*)hist)[t] = 0u;
            __syncthreads();
        } else if (pass == 1) {
            if (tid < 16) {
                unsigned c = abv0[tid], b = 0;
                for (; b < 256; ++b) {
                    unsigned nc = c + hist[tid][b];
                    if (nc >= (unsigned)KSEL) break;
                    c = nc;
                }
                th16[tid] = (cb[tid] << 8) | b;
            }
            __syncthreads();
        }
    }
    __syncthreads();

    if (tid < 16) {
        const int r = tid;
        const int have = (int)cntA[r];
        const int need = KSEL - have;
        for (int t = 0; t < need; ++t) {
            outIdx[r][have + t] = bIdx[r][t];
            outVal[r][have + t] = bVal[r][t];
        }
        // selection sort, descending dot == ascending distance (top_k order)
        for (int a = 0; a < KSEL - 1; ++a) {
            int best = a; float bv = outVal[r][a];
            for (int b = a + 1; b < KSEL; ++b) {
                float v = outVal[r][b];
                if (v > bv) { bv = v; best = b; }
            }
            if (best != a) {
                float tv = outVal[r][a]; outVal[r][a] = outVal[r][best]; outVal[r][best] = tv;
                int   ti = outIdx[r][a]; outIdx[r][a] = outIdx[r][best]; outIdx[r][best] = ti;
            }
        }
    }
    __syncthreads();

    for (int t = tid; t < 16 * KSEL; t += 256) {
        const int r = t >> 6, k = t & 63;
        nbr[(size_t)(i0 + r) * KSEL + k]  = outIdx[r][k];
        dotv[(size_t)(i0 + r) * KSEL + k] = outVal[r][k];
    }
}

// ---------------------------------------------------------------------------
// Kernel C: emit edge_index / y / mask / edge_features. One block per center.
// ---------------------------------------------------------------------------
__global__ __launch_bounds__(256) void edges_kernel(
    const float* __restrict__ xcat, const int* __restrict__ nbr,
    const float* __restrict__ dotv, const int* __restrict__ pid,
    float* __restrict__ out)
{
    __shared__ float xc[22];
    __shared__ int   sSrc[KSEL];
    __shared__ float sMask[KSEL];

    const int i   = blockIdx.x;
    const int tid = threadIdx.x;

    if (tid < 22) xc[tid] = xcat[(size_t)i * 22 + tid];

    if (tid < KSEL) {
        const int k = tid;
        const size_t e = (size_t)i * KSEL + k;
        const int src = nbr[e];
        const float d = dotv[e];
        const float dist = sqrtf(fmaxf(0.0f, 2.0f - 2.0f * d));
        const float mk = (dist < 1.0f) ? 1.0f : 0.0f;
        const float yv = ((pid[src] == pid[i]) && (mk > 0.0f)) ? 1.0f : 0.0f;
        out[OFF_EI + e]             = (float)src;   // edge_index[0] (src)
        out[OFF_EI + (size_t)NK + e] = (float)i;    // edge_index[1] (dst)
        out[OFF_Y + e]    = yv;
        out[OFF_MASK + e] = mk;
        sSrc[k] = src; sMask[k] = mk;
    }
    __syncthreads();

    for (int t = tid; t < KSEL * 44; t += 256) {
        const int k = t / 44, f = t % 44;
        const int src = sSrc[k];
        const float mk = sMask[k];
        const int ff = (f < 22) ? f : f - 22;
        const float xs = xcat[(size_t)src * 22 + ff];
        const float xd = xc[ff];
        const float v  = (f < 22) ? (xs - xd) : (xs + xd);
        out[OFF_EF + ((size_t)i * KSEL + k) * 44 + f] = v * mk;
    }
}

// ---------------------------------------------------------------------------
extern "C" void kernel_launch(void* const* d_in, const int* in_sizes, int n_in,
                              void* d_out, int out_size, void* d_ws, size_t ws_size,
                              hipStream_t stream) {
    const float* x   = (const float*)d_in[0];
    const int*   pid = (const int*)d_in[1];
    const float* W1  = (const float*)d_in[2];
    const float* b1  = (const float*)d_in[3];
    const float* W2  = (const float*)d_in[4];
    const float* b2  = (const float*)d_in[5];
    float* out = (float*)d_out;

    char* ws = (char*)d_ws;
    float* H    = (float*)ws;                                   // N*8 f32 = 512 KB
    int*   nbr  = (int*)(ws + (size_t)NN * 8 * 4);              // N*64 i32 = 4 MB
    float* dotv = (float*)(ws + (size_t)NN * 8 * 4 + (size_t)NN * KSEL * 4); // 4 MB

    embed_kernel<<<NN / 256, 256, 0, stream>>>(x, W1, b1, W2, b2, H, out);
    knn_kernel<<<NN / 16, 256, 0, stream>>>(H, nbr, dotv);
    edges_kernel<<<NN, 256, 0, stream>>>(out + OFF_XCAT, nbr, dotv, pid, out);
}